// PennesHPM__55190329754315
// MI455X (gfx1250) — compile-verified
//
#include <hip/hip_runtime.h>
#include <hip/hip_bf16.h>
#include <stdint.h>

// Pennes bio-heat pointwise map: HBM-bandwidth bound (~178 MB @ 23.3 TB/s).
// CDNA5 strategy: TDM tensor_load_to_lds DMA for the 36 B/row derivative
// stream (one descriptor per 36 KB tile, TENSORcnt-tracked), AoS-packed
// parameter grids (2x b128 gather instead of 8x b32), NT stores.

#define H_GRID 640
#define W_GRID 480
#define CHUNK 1024          // rows staged per block: 1024 * 36 B = 36 KB LDS
#define THREADS 256
#define ROWF 9              // floats per derivatives row

typedef unsigned int u32x4 __attribute__((ext_vector_type(4)));
typedef int          i32x4 __attribute__((ext_vector_type(4)));
typedef int          i32x8 __attribute__((ext_vector_type(8)));

#if defined(__has_builtin)
#if __has_builtin(__builtin_amdgcn_tensor_load_to_lds)
#define USE_TDM 1
#endif
#endif

__device__ __forceinline__ float pennes_point(float t, float u, float uxx, float uyy,
                                              float a1, float a2, float a3, float a4,
                                              float a5, float a6, float a7, float a9)
{
    // C1=0.12, C2=1.0, C3=0.003, U_BLOOD=37, U_AMBIENT=21
    float conv  = 0.12f * fmaxf(a5, 0.0f) * (uxx + uyy);
    float perf  = ((uxx + uxx) < -0.5f) ? (fmaxf(a1, 0.0f) * (37.0f - u)) : 0.0f;
    float met   = 0.003f * fmaxf(a4, 0.0f) * __expf((u - 37.0f) * 0.1f);
    float resp  = a2 * __sinf(0.62831853071795864f * t + a3);   // 2*pi*0.1
    float heart = a6 * __sinf(1.57079632679489662f * t + a7);   // 2*pi*0.25
    float cool  = fmaxf(a9, 0.0f) * (21.0f - u);
    return conv + perf + met + resp + heart + cool;
}

template<bool PACKED>
__global__ __launch_bounds__(THREADS)
void pennes_main(const float* __restrict__ deriv,
                 const float4* __restrict__ pk,
                 const float* __restrict__ g1, const float* __restrict__ g2,
                 const float* __restrict__ g3, const float* __restrict__ g4,
                 const float* __restrict__ g5, const float* __restrict__ g6,
                 const float* __restrict__ g7, const float* __restrict__ g9,
                 float* __restrict__ out, int n)
{
    __shared__ float sbuf[CHUNK * ROWF];                    // 36 KB -> ~8 blocks/WGP
    const long long e0 = (long long)blockIdx.x * CHUNK;
    long long rem = (long long)n - e0;
    if (rem <= 0) return;
    const int rows = (rem < (long long)CHUNK) ? (int)rem : CHUNK;

#if USE_TDM
    // ---- CDNA5 Tensor Data Mover: one D# DMAs the whole tile global->LDS ----
    // 1-D tile of rows*9 f32 elements (<= 9216, fits 16-bit tile_dim0).
    if (threadIdx.x < 32) {                                  // wave 0 only (TDM ignores EXEC)
        const uint64_t ga   = (uint64_t)(uintptr_t)(deriv + e0 * ROWF);
        const uint32_t lds0 = (uint32_t)(uintptr_t)sbuf;     // low 32 bits = LDS byte offset
        const uint32_t ne   = (uint32_t)rows * ROWF;         // elements (4 B each)

        u32x4 d0;                                            // D# group 0
        d0.x = 1u;                                           // count=1, user mode, no gather
        d0.y = lds0;                                         // lds_addr[31:0]
        d0.z = (uint32_t)ga;                                 // global_addr[31:0]  (bits 95:64)
        d0.w = (uint32_t)(ga >> 32) | 0x80000000u;           // ga[56:32] | type=2 (bits 127:126)

        i32x8 d1;                                            // D# group 1
        d1.s0 = (int)(2u << 16);                             // wg_mask=0, data_size=2 (4B)
        d1.s1 = (int)((ne & 0xFFFFu) << 16);                 // tensor_dim0[15:0]  (bits 63:48)
        d1.s2 = (int)(ne >> 16);                             // tensor_dim0[31:16] (bits 79:64), tensor_dim1=0
        d1.s3 = (int)((ne & 0xFFFFu) << 16);                 // tile_dim0 (bits 127:112)
        d1.s4 = 0;                                           // tile_dim1=0, tile_dim2=0 (1-D)
        d1.s5 = (int)ne;                                     // tensor_dim0_stride[31:0]
        d1.s6 = 0;                                           // stride0[47:32]=0, stride1[15:0]=0
        d1.s7 = 0;                                           // tensor_dim1_stride[47:16]=0

        i32x4 z4 = {};                                       // groups 2/3 unused (1-D tensor)
        i32x8 z8 = {};
#if __clang_major__ >= 23
        __builtin_amdgcn_tensor_load_to_lds(d0, d1, z4, z4, z8, 0);   // TENSORcnt++
#else
        __builtin_amdgcn_tensor_load_to_lds(d0, d1, z4, z4, 0);       // TENSORcnt++
#endif
        __builtin_amdgcn_s_wait_tensorcnt(0);                // tile resident in LDS
    }
    __syncthreads();                                         // publish to all 8 waves
#else
    // ---- fallback: ASYNCcnt global->LDS copy (known-good on both toolchains) ----
    {
        const char* gsrc = (const char*)(deriv + e0 * ROWF);
        const uint32_t lds0 = (uint32_t)(uintptr_t)sbuf;
        const int pieces = (rows * 36 + 15) >> 4;            // 16-byte pieces
        for (int p = threadIdx.x; p < pieces; p += THREADS) {
            uint32_t la = lds0 + ((uint32_t)p << 4);
            uint64_t gaddr = (uint64_t)(uintptr_t)(gsrc + ((size_t)p << 4));
            asm volatile("global_load_async_to_lds_b128 %0, %1, off"
                         :: "v"(la), "v"(gaddr) : "memory");
        }
        asm volatile("s_wait_asynccnt 0" ::: "memory");
    }
    __syncthreads();
#endif

    // ---- compute: LDS reads (bank-conflict-free: 9 coprime with 64) ----
    for (int r = threadIdx.x; r < rows; r += THREADS) {
        const float* row = sbuf + r * ROWF;
        float t   = row[2];
        int   xi  = (int)row[3];
        int   yi  = (int)row[4];
        float u   = row[5];
        float uxx = row[6];
        float uyy = row[7];
        int idx = xi * W_GRID + yi;

        float a1, a2, a3, a4, a5, a6, a7, a9;
        if (PACKED) {
            float4 p0 = pk[2 * idx + 0];                     // one 128B line holds all 8
            float4 p1 = pk[2 * idx + 1];
            a1 = p0.x; a2 = p0.y; a3 = p0.z; a4 = p0.w;
            a5 = p1.x; a6 = p1.y; a7 = p1.z; a9 = p1.w;
        } else {
            a1 = g1[idx]; a2 = g2[idx]; a3 = g3[idx]; a4 = g4[idx];
            a5 = g5[idx]; a6 = g6[idx]; a7 = g7[idx]; a9 = g9[idx];
        }
        float v = pennes_point(t, u, uxx, uyy, a1, a2, a3, a4, a5, a6, a7, a9);
        __builtin_nontemporal_store(v, out + e0 + r);        // don't evict gather tables
    }
}

__global__ __launch_bounds__(THREADS)
void pack_grids(const float* __restrict__ g1, const float* __restrict__ g2,
                const float* __restrict__ g3, const float* __restrict__ g4,
                const float* __restrict__ g5, const float* __restrict__ g6,
                const float* __restrict__ g7, const float* __restrict__ g9,
                float4* __restrict__ pk, int cells)
{
    int i = blockIdx.x * THREADS + threadIdx.x;
    if (i < cells) {
        pk[2 * i + 0] = make_float4(g1[i], g2[i], g3[i], g4[i]);
        pk[2 * i + 1] = make_float4(g5[i], g6[i], g7[i], g9[i]);
    }
}

extern "C" void kernel_launch(void* const* d_in, const int* in_sizes, int n_in,
                              void* d_out, int out_size, void* d_ws, size_t ws_size,
                              hipStream_t stream) {
    const float* deriv = (const float*)d_in[0];
    const float* g1 = (const float*)d_in[1];
    const float* g2 = (const float*)d_in[2];
    const float* g3 = (const float*)d_in[3];
    const float* g4 = (const float*)d_in[4];
    const float* g5 = (const float*)d_in[5];
    const float* g6 = (const float*)d_in[6];
    const float* g7 = (const float*)d_in[7];
    const float* g9 = (const float*)d_in[8];
    float* out = (float*)d_out;

    const int n = in_sizes[0] / ROWF;                // 4,194,304
    const int cells = H_GRID * W_GRID;               // 307,200
    const size_t pack_bytes = (size_t)cells * 8 * sizeof(float);   // 9.8 MB
    const int n_chunks = (n + CHUNK - 1) / CHUNK;    // 4096 blocks

    const bool packed = (d_ws != nullptr) && (ws_size >= pack_bytes);
    float4* pk = (float4*)d_ws;

    if (packed) {
        pack_grids<<<(cells + THREADS - 1) / THREADS, THREADS, 0, stream>>>(
            g1, g2, g3, g4, g5, g6, g7, g9, pk, cells);
        pennes_main<true><<<n_chunks, THREADS, 0, stream>>>(
            deriv, pk, g1, g2, g3, g4, g5, g6, g7, g9, out, n);
    } else {
        pennes_main<false><<<n_chunks, THREADS, 0, stream>>>(
            deriv, nullptr, g1, g2, g3, g4, g5, g6, g7, g9, out, n);
    }
}